// STGNP_88708254532167
// MI455X (gfx1250) — compile-verified
//
#include <hip/hip_runtime.h>
#include <math.h>

// ---------------------------------------------------------------------------
// Problem constants (match reference)
// ---------------------------------------------------------------------------
#define BB   8
#define RR   128
#define TT   64
#define FIN  32
#define LL   64
#define DD   64
#define HH   128
#define OUTD 32
#define ESN  2048
#define DTC  0.1f

typedef __attribute__((ext_vector_type(2))) float v2f;
typedef __attribute__((ext_vector_type(8))) float v8f;

#if __has_builtin(__builtin_amdgcn_wmma_f32_16x16x4_f32)
#define HAVE_WMMA_F32 1
#endif

#define DEV __device__ __forceinline__

// ---------------------------------------------------------------------------
// Fast activations: one v_exp_f32 + one v_rcp_f32 instead of IEEE divide
// ---------------------------------------------------------------------------
DEV float fast_rcp(float x) { return __builtin_amdgcn_rcpf(x); }
DEV float sigmoid_f(float z) { return fast_rcp(1.0f + __expf(-z)); }
DEV float silu_f(float z) { return z * sigmoid_f(z); }
DEV float tanh_f(float z) {
  // tanh(z) = 1 - 2/(exp(2z)+1); saturates correctly for |z| large
  return 1.0f - 2.0f * fast_rcp(__expf(2.0f * z) + 1.0f);
}

// ---------------------------------------------------------------------------
// 16x16 output tile GEMM: acc += A[16 x K] * B[K x 16], full f32 via
// V_WMMA_F32_16X16X4_F32.  A row-major (stride lda), B row-major (stride ldb),
// both in LDS.  Whole wave participates (EXEC all ones).
// A frag: lanes 0-15 hold K=k,k+1; lanes 16-31 hold K=k+2,k+3 (M=lane&15)
// B frag: mirrored (N=lane&15)
// C frag: VGPR i -> M = i + 8*(lane>>4), N = lane&15
// ---------------------------------------------------------------------------
DEV v8f tile_gemm(const float* __restrict__ A, int lda,
                  const float* __restrict__ B, int ldb,
                  int K, v8f acc) {
  const int lane = threadIdx.x & 31;
  const int r16  = lane & 15;
  const int hf   = lane >> 4;
#ifdef HAVE_WMMA_F32
  for (int k0 = 0; k0 < K; k0 += 4) {
    const int ka = k0 + 2 * hf;
    v2f a, b;
    a.x = A[r16 * lda + ka];
    a.y = A[r16 * lda + ka + 1];
    b.x = B[ka * ldb + r16];
    b.y = B[(ka + 1) * ldb + r16];
    acc = __builtin_amdgcn_wmma_f32_16x16x4_f32(
        /*neg_a=*/false, a, /*neg_b=*/false, b,
        /*c_mod=*/(short)0, acc, /*reuse_a=*/false, /*reuse_b=*/false);
  }
#else
  for (int i = 0; i < 8; ++i) {
    const int m = i + 8 * hf;
    float s = acc[i];
    for (int k = 0; k < K; ++k) s += A[m * lda + k] * B[k * ldb + r16];
    acc[i] = s;
  }
#endif
  return acc;
}

// ---------------------------------------------------------------------------
// Kernel 1: encoder.  One WG per (b,r).  X (64 x 36, padded) -> H1 -> H2,
// then mean over t -> hp[br][128].
// ---------------------------------------------------------------------------
__global__ void k_encoder(const float* __restrict__ nf,
                          const float* __restrict__ We1, const float* __restrict__ be1,
                          const float* __restrict__ We2, const float* __restrict__ be2,
                          float* __restrict__ hp) {
  extern __shared__ float sm[];
  float* Xt = sm;                    // 64*36
  float* W1 = Xt + 64 * 36;          // 36*128
  float* H1 = W1 + 36 * 128;         // 64*128
  float* W2 = H1 + 64 * 128;         // 128*128
  const int br = blockIdx.x;
  const int tid = threadIdx.x, nthr = blockDim.x;
  const int wid = tid >> 5, lane = tid & 31, col = lane & 15, hf = lane >> 4;

  __builtin_prefetch(We2, 0, 1);
  __builtin_prefetch(We1, 0, 1);

  for (int i = tid; i < 64 * 36; i += nthr) {
    int t = i / 36, c = i % 36;
    float v;
    if (c < FIN)       v = nf[(size_t)br * TT * FIN + t * FIN + c];
    else if (c == FIN) v = (float)t / (float)TT;
    else               v = 0.0f;
    Xt[i] = v;
  }
  for (int i = tid; i < 36 * 128; i += nthr) {
    int k = i / 128, n = i % 128;
    W1[i] = (k < FIN + 1) ? We1[k * 128 + n] : 0.0f;
  }
  for (int i = tid; i < 128 * 128; i += nthr) W2[i] = We2[i];
  __syncthreads();

  // H1 = silu(X @ W1 + b1): 4x8 tiles, 8 waves
  for (int tile = wid; tile < 32; tile += 8) {
    int mt = tile >> 3, nt = tile & 7;
    v8f acc = {0, 0, 0, 0, 0, 0, 0, 0};
    acc = tile_gemm(Xt + mt * 16 * 36, 36, W1 + nt * 16, 128, 36, acc);
    int n = nt * 16 + col;
    float bias = be1[n];
    for (int i = 0; i < 8; ++i) {
      int m = mt * 16 + i + 8 * hf;
      H1[m * 128 + n] = silu_f(acc[i] + bias);
    }
  }
  __syncthreads();

  // H2 = silu(H1 @ W2 + b2) written back in-place (compute-all then store)
  v8f accs[4];
  for (int j = 0; j < 4; ++j) {
    int tile = wid + j * 8;
    int mt = tile >> 3, nt = tile & 7;
    v8f a = {0, 0, 0, 0, 0, 0, 0, 0};
    accs[j] = tile_gemm(H1 + mt * 16 * 128, 128, W2 + nt * 16, 128, 128, a);
  }
  __syncthreads();
  for (int j = 0; j < 4; ++j) {
    int tile = wid + j * 8;
    int mt = tile >> 3, nt = tile & 7;
    int n = nt * 16 + col;
    float bias = be2[n];
    for (int i = 0; i < 8; ++i) {
      int m = mt * 16 + i + 8 * hf;
      H1[m * 128 + n] = silu_f(accs[j][i] + bias);
    }
  }
  __syncthreads();

  for (int n = tid; n < 128; n += nthr) {
    float s = 0.0f;
    for (int t = 0; t < TT; ++t) s += H1[t * 128 + n];
    hp[(size_t)br * HH + n] = s * (1.0f / (float)TT);
  }
}

// ---------------------------------------------------------------------------
// Kernel 2: latent head.  One WG per 16 rows of (B*R).
// v = mu + (0.1 + 0.9*sigmoid(hp@Wsig)) * eps ; Lz/Dz split ; dyn = Dz@Wdyn
// ---------------------------------------------------------------------------
__global__ void k_latent(const float* __restrict__ hp, const float* __restrict__ eps,
                         const float* __restrict__ Wmu, const float* __restrict__ Wsig,
                         const float* __restrict__ Wdyn,
                         float* __restrict__ Lz, float* __restrict__ Dz,
                         float* __restrict__ dyn) {
  extern __shared__ float sm[];
  float* HPt = sm;                    // 16*128
  float* WM  = HPt + 16 * 128;        // 128*128
  float* WS  = WM + 128 * 128;        // 128*128
  float* V   = WS + 128 * 128;        // 16*128
  float* WD  = V + 16 * 128;          // 64*64
  const int rt = blockIdx.x;          // 0..63
  const int tid = threadIdx.x, nthr = blockDim.x;
  const int wid = tid >> 5, lane = tid & 31, col = lane & 15, hf = lane >> 4;

  __builtin_prefetch(Wmu, 0, 1);
  __builtin_prefetch(Wsig, 0, 1);

  for (int i = tid; i < 16 * 128; i += nthr) HPt[i] = hp[(size_t)rt * 16 * 128 + i];
  for (int i = tid; i < 128 * 128; i += nthr) { WM[i] = Wmu[i]; WS[i] = Wsig[i]; }
  for (int i = tid; i < 64 * 64; i += nthr) WD[i] = Wdyn[i];
  __syncthreads();

  {
    const int nt = wid;   // 8 waves, 8 column tiles
    v8f am = {0, 0, 0, 0, 0, 0, 0, 0};
    v8f as = {0, 0, 0, 0, 0, 0, 0, 0};
    am = tile_gemm(HPt, 128, WM + nt * 16, 128, 128, am);
    as = tile_gemm(HPt, 128, WS + nt * 16, 128, 128, as);
    const int n = nt * 16 + col;
    for (int i = 0; i < 8; ++i) {
      int m = i + 8 * hf;
      size_t row = (size_t)rt * 16 + m;
      float sg = 0.1f + 0.9f * sigmoid_f(as[i]);
      float v = am[i] + sg * eps[row * 128 + n];
      V[m * 128 + n] = v;
      if (n < LL) Lz[row * LL + n] = v;
      else        Dz[row * DD + (n - LL)] = v;
    }
  }
  __syncthreads();

  if (wid < 4) {
    const int nt = wid;
    v8f a = {0, 0, 0, 0, 0, 0, 0, 0};
    a = tile_gemm(V + 64, 128, WD + nt * 16, 64, 64, a);
    const int n = nt * 16 + col;
    for (int i = 0; i < 8; ++i) {
      int m = i + 8 * hf;
      dyn[((size_t)rt * 16 + m) * LL + n] = a[i];
    }
  }
}

// ---------------------------------------------------------------------------
// Kernel 3: RK4 ODE scan.  One WG per batch; whole state resident in LDS.
// ---------------------------------------------------------------------------
DEV void ode_f(const float* __restrict__ in, float* __restrict__ out,
               const float* __restrict__ WM, const float* __restrict__ WSf,
               const float* __restrict__ dynS, const float* __restrict__ bode,
               const float* __restrict__ spw_b,
               const int* __restrict__ src, const int* __restrict__ dst,
               int tid, int nthr) {
  const int wid = tid >> 5, lane = tid & 31, col = lane & 15, hf = lane >> 4;
  const int nw = nthr >> 5;
  // agg (in `out`) = scatter-add of weighted source rows
  for (int i = tid; i < RR * LL; i += nthr) out[i] = 0.0f;
  __syncthreads();
  // one edge per wave: lane l handles elements l and l+32 (bank-spread, no
  // intra-wave conflicts; coalesced row reads)
  for (int e = wid; e < ESN; e += nw) {
    const float w = spw_b[e];
    const float* xr = in + src[e] * LL;
    float* ar = out + dst[e] * LL;
    atomicAdd(&ar[lane], w * xr[lane]);
    atomicAdd(&ar[lane + 32], w * xr[lane + 32]);
  }
  __syncthreads();
  // tanh(agg@Wmsg + in@Wself + dyn + b): 8x4 tiles, 16 waves -> 2 each
  v8f acc[2];
  for (int j = 0; j < 2; ++j) {
    int tile = wid + j * 16;
    int mt = tile >> 2, nt = tile & 3;
    v8f a = {0, 0, 0, 0, 0, 0, 0, 0};
    a = tile_gemm(out + mt * 16 * LL, LL, WM + nt * 16, LL, LL, a);
    a = tile_gemm(in + mt * 16 * LL, LL, WSf + nt * 16, LL, LL, a);
    acc[j] = a;
  }
  __syncthreads();   // all reads of `out` (agg) done before overwrite
  for (int j = 0; j < 2; ++j) {
    int tile = wid + j * 16;
    int mt = tile >> 2, nt = tile & 3;
    int n = nt * 16 + col;
    for (int i = 0; i < 8; ++i) {
      int m = mt * 16 + i + 8 * hf;
      out[m * LL + n] = tanh_f(acc[j][i] + dynS[m * LL + n] + bode[n]);
    }
  }
  __syncthreads();
}

__global__ void k_ode(const float* __restrict__ Lz, const float* __restrict__ dyn_g,
                      const float* __restrict__ Wmsg_g, const float* __restrict__ Wself_g,
                      const float* __restrict__ bode_g, const float* __restrict__ spw,
                      const int* __restrict__ src, const int* __restrict__ dst,
                      float* __restrict__ traj) {
  extern __shared__ float sm[];
  float* xs   = sm;                 // 128*64
  float* xtmp = xs + RR * LL;
  float* kacc = xtmp + RR * LL;
  float* kbuf = kacc + RR * LL;
  float* dynS = kbuf + RR * LL;
  float* WM   = dynS + RR * LL;     // 64*64
  float* WSf  = WM + LL * LL;       // 64*64
  float* bode = WSf + LL * LL;      // 64
  const int b = blockIdx.x;
  const int tid = threadIdx.x, nthr = blockDim.x;
  const float* spw_b = spw + (size_t)b * ESN;

  __builtin_prefetch(spw_b, 0, 1);
  __builtin_prefetch(src, 0, 1);
  __builtin_prefetch(dst, 0, 1);

  for (int i = tid; i < RR * LL; i += nthr) {
    xs[i]   = Lz[(size_t)b * RR * LL + i];
    dynS[i] = dyn_g[(size_t)b * RR * LL + i];
  }
  for (int i = tid; i < LL * LL; i += nthr) { WM[i] = Wmsg_g[i]; WSf[i] = Wself_g[i]; }
  for (int i = tid; i < LL; i += nthr) bode[i] = bode_g[i];
  __syncthreads();

  const float dt = DTC;
  for (int s = 0; s < TT; ++s) {
    ode_f(xs, kbuf, WM, WSf, dynS, bode, spw_b, src, dst, tid, nthr);        // k1
    for (int i = tid; i < RR * LL; i += nthr) {
      float k1 = kbuf[i];
      kacc[i] = k1;
      xtmp[i] = xs[i] + 0.5f * dt * k1;
    }
    __syncthreads();
    ode_f(xtmp, kbuf, WM, WSf, dynS, bode, spw_b, src, dst, tid, nthr);      // k2
    for (int i = tid; i < RR * LL; i += nthr) {
      float k2 = kbuf[i];
      kacc[i] += 2.0f * k2;
      xtmp[i] = xs[i] + 0.5f * dt * k2;
    }
    __syncthreads();
    ode_f(xtmp, kbuf, WM, WSf, dynS, bode, spw_b, src, dst, tid, nthr);      // k3
    for (int i = tid; i < RR * LL; i += nthr) {
      float k3 = kbuf[i];
      kacc[i] += 2.0f * k3;
      xtmp[i] = xs[i] + dt * k3;
    }
    __syncthreads();
    ode_f(xtmp, kbuf, WM, WSf, dynS, bode, spw_b, src, dst, tid, nthr);      // k4
    for (int i = tid; i < RR * LL; i += nthr)
      xs[i] += (dt / 6.0f) * (kacc[i] + kbuf[i]);
    __syncthreads();
    for (int i = tid; i < RR * LL; i += nthr) {
      int r = i >> 6, l = i & 63;
      traj[(((size_t)b * RR + r) * TT + s) * LL + l] = xs[i];
    }
  }
}

// ---------------------------------------------------------------------------
// Kernel 4: decoder.  One WG per (b,r).  one-hot(r) and t-feature folded in
// as rank-1 row corrections of W_d1.
// ---------------------------------------------------------------------------
__global__ void k_decoder(const float* __restrict__ traj, const float* __restrict__ Dz,
                          const float* __restrict__ Wd1, const float* __restrict__ bd1,
                          const float* __restrict__ Wd2, const float* __restrict__ bd2,
                          float* __restrict__ out) {
  extern __shared__ float sm[];
  float* A    = sm;                  // 64*128 (traj | Dz bcast)
  float* W1   = A + 64 * 128;        // 128*128 (rows 0..127 of W_d1)
  float* Hd   = W1 + 128 * 128;      // 64*128
  float* W2   = Hd + 64 * 128;       // 128*32
  float* trow = W2 + 128 * 32;       // 128  (W_d1 row 128: t coeff)
  float* rrow = trow + 128;          // 128  (W_d1 row 129+r)
  const int br = blockIdx.x;
  const int r = br & (RR - 1);
  const int tid = threadIdx.x, nthr = blockDim.x;
  const int wid = tid >> 5, lane = tid & 31, col = lane & 15, hf = lane >> 4;

  __builtin_prefetch(Wd1, 0, 1);
  __builtin_prefetch(Wd2, 0, 1);

  for (int i = tid; i < 64 * 64; i += nthr) {
    int t = i >> 6, l = i & 63;
    A[t * 128 + l] = traj[((size_t)br * TT + t) * LL + l];
  }
  for (int i = tid; i < 64 * 64; i += nthr) {
    int t = i >> 6, d = i & 63;
    A[t * 128 + 64 + d] = Dz[(size_t)br * DD + d];
  }
  for (int i = tid; i < 128 * 128; i += nthr) W1[i] = Wd1[i];
  for (int i = tid; i < 128 * 32; i += nthr) W2[i] = Wd2[i];
  for (int i = tid; i < 128; i += nthr) {
    trow[i] = Wd1[128 * 128 + i];
    rrow[i] = Wd1[(129 + r) * 128 + i];
  }
  __syncthreads();

  for (int tile = wid; tile < 32; tile += 8) {
    int mt = tile >> 3, nt = tile & 7;
    v8f a = {0, 0, 0, 0, 0, 0, 0, 0};
    a = tile_gemm(A + mt * 16 * 128, 128, W1 + nt * 16, 128, 128, a);
    int n = nt * 16 + col;
    float bias = bd1[n];
    for (int i = 0; i < 8; ++i) {
      int m = mt * 16 + i + 8 * hf;   // m == t
      float tf = (float)m / (float)TT;
      Hd[m * 128 + n] = silu_f(a[i] + tf * trow[n] + rrow[n] + bias);
    }
  }
  __syncthreads();

  {
    int tile = wid;                 // 8 waves, 4x2 tiles
    int mt = tile >> 1, nt = tile & 1;
    v8f a = {0, 0, 0, 0, 0, 0, 0, 0};
    a = tile_gemm(Hd + mt * 16 * 128, 128, W2 + nt * 16, OUTD, 128, a);
    int n = nt * 16 + col;
    float bias = bd2[n];
    for (int i = 0; i < 8; ++i) {
      int m = mt * 16 + i + 8 * hf;
      out[((size_t)br * TT + m) * OUTD + n] = a[i] + bias;
    }
  }
}

// ---------------------------------------------------------------------------
// Host launcher
// ---------------------------------------------------------------------------
extern "C" void kernel_launch(void* const* d_in, const int* in_sizes, int n_in,
                              void* d_out, int out_size, void* d_ws, size_t ws_size,
                              hipStream_t stream) {
  const float* nf    = (const float*)d_in[0];
  const float* eps   = (const float*)d_in[1];
  const float* spw   = (const float*)d_in[2];
  const int*   src   = (const int*)d_in[3];
  const int*   dst   = (const int*)d_in[4];
  const float* We1   = (const float*)d_in[5];
  const float* be1   = (const float*)d_in[6];
  const float* We2   = (const float*)d_in[7];
  const float* be2   = (const float*)d_in[8];
  const float* Wmu   = (const float*)d_in[9];
  const float* Wsig  = (const float*)d_in[10];
  const float* Wmsg  = (const float*)d_in[11];
  const float* Wself = (const float*)d_in[12];
  const float* Wdyn  = (const float*)d_in[13];
  const float* bode  = (const float*)d_in[14];
  const float* Wd1   = (const float*)d_in[15];
  const float* bd1   = (const float*)d_in[16];
  const float* Wd2   = (const float*)d_in[17];
  const float* bd2   = (const float*)d_in[18];
  float* out = (float*)d_out;

  float* ws   = (float*)d_ws;
  float* hp   = ws;                          // B*R*H     = 131072
  float* Lz   = hp + (size_t)BB * RR * HH;   // B*R*L     = 65536
  float* Dz   = Lz + (size_t)BB * RR * LL;   // B*R*D     = 65536
  float* dyn  = Dz + (size_t)BB * RR * DD;   // B*R*L     = 65536
  float* traj = dyn + (size_t)BB * RR * LL;  // B*R*T*L   = 4194304

  const size_t lds_enc = (size_t)(64 * 36 + 36 * 128 + 64 * 128 + 128 * 128) * 4;
  const size_t lds_lat = (size_t)(16 * 128 + 128 * 128 + 128 * 128 + 16 * 128 + 64 * 64) * 4;
  const size_t lds_ode = (size_t)(5 * RR * LL + 2 * LL * LL + LL) * 4;
  const size_t lds_dec = (size_t)(64 * 128 + 128 * 128 + 64 * 128 + 128 * 32 + 256) * 4;

  k_encoder<<<BB * RR, 256, lds_enc, stream>>>(nf, We1, be1, We2, be2, hp);
  k_latent<<<(BB * RR) / 16, 256, lds_lat, stream>>>(hp, eps, Wmu, Wsig, Wdyn, Lz, Dz, dyn);
  k_ode<<<BB, 512, lds_ode, stream>>>(Lz, dyn, Wmsg, Wself, bode, spw, src, dst, traj);
  k_decoder<<<BB * RR, 256, lds_dec, stream>>>(traj, Dz, Wd1, bd1, Wd2, bd2, out);
}